// MultiHeadQuantizationV2_45492293599493
// MI455X (gfx1250) — compile-verified
//
#include <hip/hip_runtime.h>

typedef float v2f __attribute__((ext_vector_type(2)));
typedef float v4f __attribute__((ext_vector_type(4)));
typedef float v8f __attribute__((ext_vector_type(8)));

#define VQ_N 32768
#define VQ_H 8
#define VQ_D 64
#define VQ_K 512
#define CB_STRIDE 68        // padded LDS stride (floats): 4-float bank stride -> conflict-free b64 reads
#define KQ 128              // codes staged in LDS per pass (128*68*4 = 34.8 KB)
#define BLK 256
#define ROWS_PER_BLOCK 128  // 8 waves x 16 rows

// CDNA5 async memory->LDS copy (16B per lane), tracked by ASYNCcnt.
__device__ __forceinline__ void async_copy16(unsigned lds_byte_off, const void* gptr) {
  asm volatile("global_load_async_to_lds_b128 %0, %1, off"
               :: "v"(lds_byte_off), "v"((unsigned long long)(uintptr_t)gptr)
               : "memory");
}
__device__ __forceinline__ void wait_asynccnt0() {
  asm volatile("s_wait_asynccnt 0x0" ::: "memory");
}

__global__ __launch_bounds__(BLK) void vq_main(
    const float* __restrict__ embeds, const float* __restrict__ codebook,
    float* __restrict__ out_q, float* __restrict__ out_idx,
    float* __restrict__ partial)
{
  __shared__ float cb[KQ * CB_STRIDE];
  __shared__ float cnormAll[VQ_K];
  __shared__ int   idxbuf[ROWS_PER_BLOCK];
  __shared__ float errbuf[BLK / 32];

  const int h    = blockIdx.x & (VQ_H - 1);
  const int tile = blockIdx.x >> 3;
  const int rowBase = tile * ROWS_PER_BLOCK;
  const int t    = threadIdx.x;
  const int wave = t >> 5;
  const int lane = t & 31;
  const int col  = lane & 15;
  const int half = lane >> 4;

  // ---- precompute ||c||^2 for all K codes of this head (codebook is L2-hot) ----
  for (int c = t; c < VQ_K; c += BLK) {
    const float* src = codebook + ((size_t)h * VQ_K + c) * VQ_D;
    float ss = 0.f;
    #pragma unroll
    for (int d = 0; d < VQ_D; d += 4) {
      v4f v = *(const v4f*)(src + d);
      ss += v.x * v.x + v.y * v.y + v.z * v.z + v.w * v.w;
    }
    cnormAll[c] = ss;
  }

  // ---- A fragments: this wave's 16 rows of z, 16 chunks of K=4 (f32 16x16x4 layout) ----
  const int rbase = rowBase + wave * 16;
  const float* zrow = embeds + ((size_t)(rbase + col) * VQ_H + h) * VQ_D;
  v2f a[16];
  #pragma unroll
  for (int c = 0; c < 16; ++c)
    a[c] = *(const v2f*)(zrow + 4 * c + 2 * half);

  float minv[8];
  int   mini[8];
  #pragma unroll
  for (int v = 0; v < 8; ++v) { minv[v] = 3.402823e38f; mini[v] = 0; }

  for (int kq = 0; kq < VQ_K / KQ; ++kq) {
    __syncthreads();                       // previous quarter fully consumed (also covers cnormAll)
    // ---- async-stage KQ codes of head h into padded LDS (no VGPR round-trip) ----
    {
      const float* base = codebook + ((size_t)h * VQ_K + kq * KQ) * VQ_D;
      #pragma unroll
      for (int i = 0; i < (KQ * VQ_D) / 4 / BLK; ++i) {   // 2048 16B-chunks / 256 thr = 8
        int j    = t + BLK * i;
        int code = j >> 4;                 // 16 chunks per code
        int dpos = (j & 15) * 4;
        unsigned ldsoff = (unsigned)(uintptr_t)(const void*)&cb[code * CB_STRIDE + dpos];
        async_copy16(ldsoff, base + (size_t)code * VQ_D + dpos);
      }
    }
    wait_asynccnt0();                      // our async writes landed in LDS
    __syncthreads();                       // publish to all waves

    for (int kt = 0; kt < KQ / 16; ++kt) {
      const int cl = kt * 16 + col;        // code local to the staged quarter
      const float* bsrc = cb + cl * CB_STRIDE + 2 * half;
      v2f b[16];
      #pragma unroll
      for (int c = 0; c < 16; ++c)
        b[c] = *(const v2f*)(bsrc + 4 * c);
      v8f acc = {};
      #pragma unroll
      for (int c = 0; c < 16; ++c)
        acc = __builtin_amdgcn_wmma_f32_16x16x4_f32(
            /*neg_a=*/false, a[c], /*neg_b=*/false, b[c],
            /*c_mod=*/(short)0, acc, /*reuse_a=*/false, /*reuse_b=*/false);
      const int code = kq * KQ + cl;
      const float cn = cnormAll[code];
      #pragma unroll
      for (int v = 0; v < 8; ++v) {
        float s = cn - 2.0f * acc[v];      // ||c||^2 - 2 z.c  (||z||^2 constant per row)
        if (s < minv[v]) { minv[v] = s; mini[v] = code; }  // strictly-< keeps first index
      }
    }
  }

  // ---- argmin reduction across the 16 code-columns (lanes sharing a row) ----
  #pragma unroll
  for (int off = 1; off < 16; off <<= 1) {
    #pragma unroll
    for (int v = 0; v < 8; ++v) {
      float ov = __shfl_xor(minv[v], off, 32);
      int   oi = __shfl_xor(mini[v], off, 32);
      if (ov < minv[v] || (ov == minv[v] && oi < mini[v])) { minv[v] = ov; mini[v] = oi; }
    }
  }
  if (col == 0) {          // lanes 0 & 16 hold rows v+8*half of this wave's tile
    #pragma unroll
    for (int v = 0; v < 8; ++v) {
      int r = wave * 16 + half * 8 + v;
      idxbuf[r] = mini[v];
      out_idx[(size_t)(rowBase + r) * VQ_H + h] = (float)mini[v];
    }
  }
  __syncthreads();

  // ---- gather z_q, straight-through output, SSE for the loss ----
  {
    const int grow = col;                  // local row within wave tile
    const int dOff = half * 32;            // each half-wave handles 32 of D=64
    const int code = idxbuf[wave * 16 + grow];
    const float* csrc = codebook + ((size_t)h * VQ_K + code) * VQ_D + dOff;
    const size_t gr = (size_t)(rbase + grow);
    const float* zsrc = embeds + (gr * VQ_H + h) * VQ_D + dOff;
    float* qdst = out_q + (gr * VQ_H + h) * VQ_D + dOff;
    float err = 0.f;
    #pragma unroll
    for (int d = 0; d < 32; d += 4) {
      v4f cv = *(const v4f*)(csrc + d);
      v4f zv = *(const v4f*)(zsrc + d);
      v4f q  = zv + (cv - zv);             // straight-through forward value, ref rounding order
      *(v4f*)(qdst + d) = q;
      v4f e = zv - cv;
      err += e.x * e.x + e.y * e.y + e.z * e.z + e.w * e.w;
    }
    #pragma unroll
    for (int off = 1; off < 32; off <<= 1)
      err += __shfl_xor(err, off, 32);
    if (lane == 0) errbuf[wave] = err;
  }
  __syncthreads();
  if (t == 0) {
    float s = 0.f;
    #pragma unroll
    for (int w = 0; w < BLK / 32; ++w) s += errbuf[w];
    partial[blockIdx.x] = s;               // deterministic per-block partial (no float atomics)
  }
}

__global__ __launch_bounds__(256) void vq_loss(const float* __restrict__ partial,
                                               float* __restrict__ loss, int nparts)
{
  __shared__ float buf[256];
  float s = 0.f;
  for (int i = threadIdx.x; i < nparts; i += 256) s += partial[i];  // fixed order
  buf[threadIdx.x] = s;
  __syncthreads();
  for (int off = 128; off > 0; off >>= 1) {
    if (threadIdx.x < off) buf[threadIdx.x] += buf[threadIdx.x + off];
    __syncthreads();
  }
  if (threadIdx.x == 0)
    *loss = 1.25f * buf[0] / (float)(VQ_N * VQ_H * VQ_D);  // beta*commit + codebook = 1.25*mse
}

extern "C" void kernel_launch(void* const* d_in, const int* in_sizes, int n_in,
                              void* d_out, int out_size, void* d_ws, size_t ws_size,
                              hipStream_t stream)
{
  const float* embeds   = (const float*)d_in[0];
  const float* codebook = (const float*)d_in[1];
  float* out     = (float*)d_out;
  float* out_q   = out;                                    // [N,H,D]
  float* out_idx = out + (size_t)VQ_N * VQ_H * VQ_D;       // [N,H] as float
  float* loss    = out_idx + (size_t)VQ_N * VQ_H;          // scalar
  float* partial = (float*)d_ws;                           // one float per block

  const int nblocks = (VQ_N / ROWS_PER_BLOCK) * VQ_H;      // 2048
  vq_main<<<nblocks, BLK, 0, stream>>>(embeds, codebook, out_q, out_idx, partial);
  vq_loss<<<1, 256, 0, stream>>>(partial, loss, nblocks);
  (void)in_sizes; (void)n_in; (void)out_size; (void)ws_size;
}